// DeepHodgeLayer_53291954208927
// MI455X (gfx1250) — compile-verified
//
#include <hip/hip_runtime.h>
#include <math.h>

// ---------------------------------------------------------------------------
// DeepHodgeLayer forward for MI455X (gfx1250), fp32 WMMA (V_WMMA_F32_16X16X4_F32)
// Block = one batch element, 256 threads = 8 wave32s.
// Edge branch folded algebraically: M1n[s] = N[s] @ (xn @ We_s) + bias,
//   N[s] = R diag(mask) R^T + Q diag(W2[s]) Q^T + tau*S0   (all 16x16)
// with constants Q (16x560), R, S0, r1, s1 precomputed into d_ws.
// ---------------------------------------------------------------------------

typedef __attribute__((ext_vector_type(2))) float v2f;
typedef __attribute__((ext_vector_type(8))) float v8f;

#define KNODE  16
#define EDGES  120
#define TRIS   560
#define DMODEL 256
#define TAUC   1e-4f
#define LDX    260      // LDS row stride for 16x256 tiles (260 % 64 == 4 -> conflict-free A-frags)
#define LDH    1028     // LDS row stride for 16x1024 FFN hidden

// workspace layout (floats)
#define WS_Q   0
#define WS_R   (KNODE*TRIS)        // 8960
#define WS_S0  (WS_R + 256)        // 9216
#define WS_R1  (WS_S0 + 256)       // 9472
#define WS_S1  (WS_R1 + 16)        // 9488
#define WS_TOT (WS_S1 + 16)        // 9504

#define WMMA4(c,a,b) (c) = __builtin_amdgcn_wmma_f32_16x16x4_f32(false,(a),false,(b),(short)0,(c),false,false)

// A-fragment (16xK slice, K-step of 4) per ISA 7.12.2: lanes 0-15 -> K=kb,kb+1; lanes 16-31 -> kb+2,kb+3
__device__ __forceinline__ v2f frag_a(const float* src, int ld, int kb, int ln) {
  int row = ln & 15;
  int kk  = kb + ((ln >> 4) << 1);
  v2f a;
  a.x = src[row * ld + kk];
  a.y = src[row * ld + kk + 1];
  return a;
}
// B-fragment from global row-major W (Kdim x ldw), 16-col tile at coloff
__device__ __forceinline__ v2f frag_b_g(const float* W, int ldw, int kb, int coloff, int ln) {
  int col = ln & 15;
  int kk  = kb + ((ln >> 4) << 1);
  v2f b;
  b.x = W[(size_t)kk * ldw + coloff + col];
  b.y = W[(size_t)(kk + 1) * ldw + coloff + col];
  return b;
}
// B-fragment from LDS (row-major, stride ld)
__device__ __forceinline__ v2f frag_b_s(const float* src, int ld, int kb, int ln) {
  int col = ln & 15;
  int kk  = kb + ((ln >> 4) << 1);
  v2f b;
  b.x = src[kk * ld + col];
  b.y = src[(kk + 1) * ld + col];
  return b;
}

// ---------------------------------------------------------------------------
// Prelude: batch-independent constants from B1 / |B1| / triangle index arrays.
// Q[n][t] = sum_e |B1|[n,e] * B2[e,t] = |B1|[n,t_jk] - |B1|[n,t_ik] + |B1|[n,t_ij]
// ---------------------------------------------------------------------------
__global__ void hodge_prelude(const float* __restrict__ B1,
                              const float* __restrict__ absB1,
                              const int* __restrict__ t_ij,
                              const int* __restrict__ t_jk,
                              const int* __restrict__ t_ik,
                              float* __restrict__ ws) {
  int idx = blockIdx.x * blockDim.x + threadIdx.x;
  if (idx < KNODE * TRIS) {
    int n = idx / TRIS, t = idx % TRIS;
    const float* ab = absB1 + n * EDGES;
    ws[WS_Q + idx] = ab[t_jk[t]] - ab[t_ik[t]] + ab[t_ij[t]];
  } else if (idx < KNODE * TRIS + 256) {            // R = |B1| B1^T
    int r = idx - KNODE * TRIS, i = r >> 4, k = r & 15;
    float acc = 0.f;
    for (int e = 0; e < EDGES; ++e) acc += absB1[i * EDGES + e] * B1[k * EDGES + e];
    ws[WS_R + r] = acc;
  } else if (idx < KNODE * TRIS + 512) {            // S0 = |B1| |B1|^T
    int r = idx - KNODE * TRIS - 256, i = r >> 4, k = r & 15;
    float acc = 0.f;
    for (int e = 0; e < EDGES; ++e) acc += absB1[i * EDGES + e] * absB1[k * EDGES + e];
    ws[WS_S0 + r] = acc;
  } else if (idx < KNODE * TRIS + 512 + 16) {       // r1 = B1 @ 1
    int k = idx - (KNODE * TRIS + 512);
    float acc = 0.f;
    for (int e = 0; e < EDGES; ++e) acc += B1[k * EDGES + e];
    ws[WS_R1 + k] = acc;
  } else if (idx < KNODE * TRIS + 512 + 32) {       // s1 = |B1| @ 1
    int k = idx - (KNODE * TRIS + 528);
    float acc = 0.f;
    for (int e = 0; e < EDGES; ++e) acc += absB1[k * EDGES + e];
    ws[WS_S1 + k] = acc;
  }
}

// ---------------------------------------------------------------------------
// Main fused kernel: one block per batch element.
// ---------------------------------------------------------------------------
__global__ __launch_bounds__(256) void hodge_main(
    const float* __restrict__ x, const float* __restrict__ mask,
    const int* __restrict__ e_i, const int* __restrict__ e_j,
    const int* __restrict__ t_ij, const int* __restrict__ t_jk, const int* __restrict__ t_ik,
    const float* __restrict__ log_scales,
    const float* __restrict__ Wg, const float* __restrict__ bg,
    const float* __restrict__ Wv0, const float* __restrict__ bv0,
    const float* __restrict__ We, const float* __restrict__ be,
    const float* __restrict__ Wout, const float* __restrict__ bout,
    const float* __restrict__ g1, const float* __restrict__ b1,
    const float* __restrict__ g2, const float* __restrict__ b2,
    const float* __restrict__ Wf1, const float* __restrict__ bf1,
    const float* __restrict__ Wf2, const float* __restrict__ bf2,
    const float* __restrict__ ws, float* __restrict__ out) {
  extern __shared__ float smem[];
  float* xn   = smem;                 // 16*LDX  : x_norm, later x2_norm
  float* x2s  = xn + 16 * LDX;        // 16*LDX  : x2 (residual+attn)
  float* Ms   = x2s + 16 * LDX;       // 16*LDX  : per-pass M matrix
  float* hs   = Ms + 16 * LDX;        // 16*LDH  : Q staging, later FFN hidden
  float* Pm   = hs + 16 * LDH;        // 16*16
  float* Asym = Pm + 256;             // 3*16*16 : Gaussian kernel A[s]
  float* L0s  = Asym + 768;           // 3*16*16
  float* Ns   = L0s + 768;            // 3*16*16
  float* W1s  = Ns + 768;             // 3*120
  float* W2s  = W1s + 360;            // 3*560
  float* pns  = W2s + 1680;           // 16
  float* bev  = pns + 16;             // 3*16 bias_edge

  const int b  = blockIdx.x;
  const int tid = threadIdx.x;
  const int wv = tid >> 5;            // wave id 0..7
  const int ln = tid & 31;            // lane id
  const float* xb = x + (size_t)b * KNODE * DMODEL;
  const float* mb = mask + b * KNODE;

  __builtin_prefetch(Wout, 0, 2);
  __builtin_prefetch(Wf1, 0, 2);

  // ---- Stage A: layernorm(x) -> xn -------------------------------------
  for (int r = wv; r < KNODE; r += 8) {
    float s = 0.f, s2 = 0.f;
#pragma unroll
    for (int k = 0; k < 8; ++k) { float v = xb[r * DMODEL + ln + k * 32]; s += v; s2 += v * v; }
#pragma unroll
    for (int off = 16; off > 0; off >>= 1) { s += __shfl_xor(s, off, 32); s2 += __shfl_xor(s2, off, 32); }
    float mu = s * (1.f / DMODEL);
    float var = s2 * (1.f / DMODEL) - mu * mu;
    float rstd = rsqrtf(var + 1e-5f);
#pragma unroll
    for (int k = 0; k < 8; ++k) {
      int c = ln + k * 32;
      xn[r * LDX + c] = (xb[r * DMODEL + c] - mu) * rstd * g1[c] + b1[c];
    }
  }
  __syncthreads();

  // ---- Stage B: P = xn @ Wg + bg (one 16x16 tile, wave 0, WMMA) --------
  if (wv == 0) {
    v8f c = {0.f, 0.f, 0.f, 0.f, 0.f, 0.f, 0.f, 0.f};
    for (int kb = 0; kb < DMODEL; kb += 4) {
      v2f a = frag_a(xn, LDX, kb, ln);
      v2f bb = frag_b_g(Wg, 16, kb, 0, ln);
      WMMA4(c, a, bb);
    }
    int col = ln & 15, rb = (ln >> 4) * 8;
#pragma unroll
    for (int v = 0; v < 8; ++v) Pm[(rb + v) * 16 + col] = c[v] + bg[col];
  }
  __syncthreads();
  if (tid < 16) {
    float acc = 0.f;
    for (int c2 = 0; c2 < 16; ++c2) { float p = Pm[tid * 16 + c2]; acc += p * p; }
    pns[tid] = acc;
  }
  __syncthreads();

  // ---- Stage C: A[s], W1, W2, L0 ---------------------------------------
  float inv2s[3];
#pragma unroll
  for (int s2 = 0; s2 < 3; ++s2) {
    float sg = expf(log_scales[s2]);
    inv2s[s2] = 1.f / (2.f * sg * sg + 1e-8f);
  }
  {
    int i = tid >> 4, j = tid & 15;
    float dot = 0.f;
    for (int c2 = 0; c2 < 16; ++c2) dot += Pm[i * 16 + c2] * Pm[j * 16 + c2];
    float m2 = mb[i] * mb[j];
    float d2 = pns[i] + pns[j] - 2.f * dot;
    float dm = (d2 > 0.f) ? sqrtf(d2) : 0.f;
    dm *= m2;
    float dmsq = dm * dm;
#pragma unroll
    for (int s2 = 0; s2 < 3; ++s2) Asym[s2 * 256 + tid] = expf(-dmsq * inv2s[s2]) * m2;
  }
  __syncthreads();
  for (int idx = tid; idx < 3 * EDGES; idx += 256) {
    int s2 = idx / EDGES, e = idx % EDGES;
    W1s[idx] = Asym[s2 * 256 + e_i[e] * 16 + e_j[e]];
  }
  __syncthreads();
  for (int idx = tid; idx < 3 * TRIS; idx += 256) {
    int s2 = idx / TRIS, t = idx % TRIS;
    const float* w1 = W1s + s2 * EDGES;
    W2s[idx] = w1[t_ij[t]] * w1[t_jk[t]] * w1[t_ik[t]];
  }
  // L0[s] = B1 diag(W1) B1^T + tau I  ==  diag(rowsum A) - A_offdiag + tau I
  for (int idx = tid; idx < 768; idx += 256) {
    int s2 = idx >> 8, r = idx & 255, i = r >> 4, j = r & 15;
    float v;
    if (i == j) {
      v = TAUC;
      for (int m = 0; m < 16; ++m) if (m != i) v += Asym[s2 * 256 + i * 16 + m];
    } else {
      v = -Asym[s2 * 256 + i * 16 + j];
    }
    L0s[idx] = v;
  }
  // stage Q (16x560) into hs (reused later for FFN hidden)
  for (int idx = tid; idx < KNODE * TRIS; idx += 256) hs[idx] = ws[WS_Q + idx];
  __syncthreads();

  // ---- Stage D: N[s] (folded edge Laplacian) and bias_edge -------------
  const float* Rm  = ws + WS_R;
  const float* S0m = ws + WS_S0;
  for (int idx = tid; idx < 768; idx += 256) {
    int s2 = idx >> 8, r = idx & 255, i = r >> 4, j = r & 15;
    float acc = TAUC * S0m[r];
    for (int m = 0; m < 16; ++m) acc += Rm[i * 16 + m] * mb[m] * Rm[j * 16 + m];
    const float* qi = hs + i * TRIS;
    const float* qj = hs + j * TRIS;
    const float* w2 = W2s + s2 * TRIS;
    for (int t = 0; t < TRIS; ++t) acc += qi[t] * qj[t] * w2[t];
    Ns[idx] = acc;
  }
  if (tid < 48) {                       // bias_edge[s][k] = |B1| L1[s] 1
    int s2 = tid / 16, k2 = tid & 15;
    float acc = TAUC * ws[WS_S1 + k2];
    for (int m = 0; m < 16; ++m) acc += Rm[k2 * 16 + m] * mb[m] * ws[WS_R1 + m];
    const float* qk = hs + k2 * TRIS;
    const float* w2 = W2s + s2 * TRIS;
    for (int t = 0; t < TRIS; ++t) acc += qk[t] * w2[t];
    bev[tid] = acc;
  }
  __syncthreads();

  // ---- Stage E: 6 passes (branch x scale); attn accumulated in C-frags -
  v8f acc0 = {0.f, 0.f, 0.f, 0.f, 0.f, 0.f, 0.f, 0.f};
  v8f acc1 = {0.f, 0.f, 0.f, 0.f, 0.f, 0.f, 0.f, 0.f};
  const int jo0 = wv * 2, jo1 = wv * 2 + 1;   // this wave's output column tiles
  const int col = ln & 15, rb = (ln >> 4) * 8;
  for (int p = 0; p < 6; ++p) {
    int br = p / 3, s2 = p % 3;
    const float* Wbig = br ? We : Wv0;                       // (256, 768)
    const float* coef = (br ? Ns : L0s) + s2 * 256;          // 16x16
    const float* bvec = (br ? be : bv0) + s2 * DMODEL;       // bias row
#pragma unroll 1
    for (int it = 0; it < 2; ++it) {
      int jt = wv * 2 + it;
      // Y tile = xn @ Wbig[:, s*256 + jt*16 : +16]   (64 WMMA k-steps)
      v8f cy = {0.f, 0.f, 0.f, 0.f, 0.f, 0.f, 0.f, 0.f};
      for (int kb = 0; kb < DMODEL; kb += 4) {
        v2f a = frag_a(xn, LDX, kb, ln);
        v2f bb = frag_b_g(Wbig, 3 * DMODEL, kb, s2 * DMODEL + jt * 16, ln);
        WMMA4(cy, a, bb);
      }
      float* tile = Ms + jt * 16;
#pragma unroll
      for (int v = 0; v < 8; ++v) tile[(rb + v) * LDX + col] = cy[v];
      // T tile = coef(16x16) @ Y tile  (4 WMMA k-steps; intra-wave LDS RAW, DS in-order)
      v8f ct = {0.f, 0.f, 0.f, 0.f, 0.f, 0.f, 0.f, 0.f};
#pragma unroll
      for (int kb = 0; kb < 16; kb += 4) {
        v2f a = frag_a(coef, 16, kb, ln);
        v2f bb = frag_b_s(tile, LDX, kb, ln);
        WMMA4(ct, a, bb);
      }
      float bcol = bvec[jt * 16 + col];
#pragma unroll
      for (int v = 0; v < 8; ++v) {
        int row = rb + v;
        float biasval = br ? (bev[s2 * 16 + row] * bcol) : (TAUC * bcol);
        tile[row * LDX + col] = ct[v] + biasval;
      }
    }
    __syncthreads();   // full M (16x256) ready for everyone
    // attn += M @ Wout[branch*768 + s*256 : +256, :]
    const float* Wb = Wout + (size_t)(br * 768 + s2 * DMODEL) * DMODEL;
    for (int kb = 0; kb < DMODEL; kb += 4) {
      v2f a  = frag_a(Ms, LDX, kb, ln);
      v2f b0 = frag_b_g(Wb, DMODEL, kb, jo0 * 16, ln);
      WMMA4(acc0, a, b0);
      v2f b1v = frag_b_g(Wb, DMODEL, kb, jo1 * 16, ln);
      WMMA4(acc1, a, b1v);
    }
    __syncthreads();   // before next pass overwrites Ms
  }

  // x2 = residual + attn + bout  -> x2s
#pragma unroll
  for (int v = 0; v < 8; ++v) {
    int row = rb + v;
    x2s[row * LDX + jo0 * 16 + col] = xb[row * DMODEL + jo0 * 16 + col] + acc0[v] + bout[jo0 * 16 + col];
    x2s[row * LDX + jo1 * 16 + col] = xb[row * DMODEL + jo1 * 16 + col] + acc1[v] + bout[jo1 * 16 + col];
  }
  __syncthreads();

  // ---- Stage F: layernorm(x2) -> xn (reuse) ----------------------------
  for (int r = wv; r < KNODE; r += 8) {
    float s = 0.f, s2 = 0.f;
#pragma unroll
    for (int k = 0; k < 8; ++k) { float v = x2s[r * LDX + ln + k * 32]; s += v; s2 += v * v; }
#pragma unroll
    for (int off = 16; off > 0; off >>= 1) { s += __shfl_xor(s, off, 32); s2 += __shfl_xor(s2, off, 32); }
    float mu = s * (1.f / DMODEL);
    float var = s2 * (1.f / DMODEL) - mu * mu;
    float rstd = rsqrtf(var + 1e-5f);
#pragma unroll
    for (int k = 0; k < 8; ++k) {
      int c = ln + k * 32;
      xn[r * LDX + c] = (x2s[r * LDX + c] - mu) * rstd * g2[c] + b2[c];
    }
  }
  __syncthreads();

  // ---- Stage G: FFN hidden h = gelu(x2n @ Wf1 + bf1) -> hs -------------
  for (int jh = wv * 8; jh < wv * 8 + 8; ++jh) {
    v8f c = {0.f, 0.f, 0.f, 0.f, 0.f, 0.f, 0.f, 0.f};
    for (int kb = 0; kb < DMODEL; kb += 4) {
      v2f a = frag_a(xn, LDX, kb, ln);
      v2f bb = frag_b_g(Wf1, 1024, kb, jh * 16, ln);
      WMMA4(c, a, bb);
    }
    float bias = bf1[jh * 16 + col];
#pragma unroll
    for (int v = 0; v < 8; ++v) {
      float hv = c[v] + bias;
      hv = 0.5f * hv * (1.f + erff(hv * 0.70710678118654752f));   // exact gelu
      hs[(rb + v) * LDH + jh * 16 + col] = hv;
    }
  }
  __syncthreads();

  // ---- Stage H: out = x2 + h @ Wf2 + bf2 -------------------------------
  v8f o0 = {0.f, 0.f, 0.f, 0.f, 0.f, 0.f, 0.f, 0.f};
  v8f o1 = {0.f, 0.f, 0.f, 0.f, 0.f, 0.f, 0.f, 0.f};
  for (int kb = 0; kb < 1024; kb += 4) {
    v2f a  = frag_a(hs, LDH, kb, ln);
    v2f b0 = frag_b_g(Wf2, DMODEL, kb, jo0 * 16, ln);
    WMMA4(o0, a, b0);
    v2f b1v = frag_b_g(Wf2, DMODEL, kb, jo1 * 16, ln);
    WMMA4(o1, a, b1v);
  }
  float* ob = out + (size_t)b * KNODE * DMODEL;
#pragma unroll
  for (int v = 0; v < 8; ++v) {
    int row = rb + v;
    ob[row * DMODEL + jo0 * 16 + col] = x2s[row * LDX + jo0 * 16 + col] + o0[v] + bf2[jo0 * 16 + col];
    ob[row * DMODEL + jo1 * 16 + col] = x2s[row * LDX + jo1 * 16 + col] + o1[v] + bf2[jo1 * 16 + col];
  }
}

// ---------------------------------------------------------------------------
extern "C" void kernel_launch(void* const* d_in, const int* in_sizes, int n_in,
                              void* d_out, int out_size, void* d_ws, size_t ws_size,
                              hipStream_t stream) {
  (void)in_sizes; (void)n_in; (void)out_size; (void)ws_size;
  const float* x    = (const float*)d_in[0];
  const float* mask = (const float*)d_in[1];
  const float* B1   = (const float*)d_in[2];
  const float* B2   = (const float*)d_in[3]; (void)B2; // folded via t_* index arrays
  const float* aB1  = (const float*)d_in[4];
  const int* e_i    = (const int*)d_in[5];
  const int* e_j    = (const int*)d_in[6];
  const int* t_ij   = (const int*)d_in[7];
  const int* t_jk   = (const int*)d_in[8];
  const int* t_ik   = (const int*)d_in[9];
  const float* lsc  = (const float*)d_in[10];
  const float* Wg   = (const float*)d_in[11];
  const float* bg   = (const float*)d_in[12];
  const float* Wv0  = (const float*)d_in[13];
  const float* bv0  = (const float*)d_in[14];
  const float* We   = (const float*)d_in[15];
  const float* be   = (const float*)d_in[16];
  const float* Wout = (const float*)d_in[17];
  const float* bout = (const float*)d_in[18];
  const float* g1   = (const float*)d_in[19];
  const float* b1   = (const float*)d_in[20];
  const float* g2   = (const float*)d_in[21];
  const float* b2   = (const float*)d_in[22];
  const float* Wf1  = (const float*)d_in[23];
  const float* bf1  = (const float*)d_in[24];
  const float* Wf2  = (const float*)d_in[25];
  const float* bf2  = (const float*)d_in[26];
  float* ws  = (float*)d_ws;
  float* outp = (float*)d_out;

  hodge_prelude<<<(WS_TOT + 255) / 256, 256, 0, stream>>>(B1, aB1, t_ij, t_jk, t_ik, ws);

  size_t smem = (size_t)(3 * 16 * LDX + 16 * LDH + 256 + 3 * 768 + 360 + 1680 + 16 + 48) * sizeof(float);
  static_assert(16 * LDH >= KNODE * TRIS, "Q staging must fit in hs");
  hipFuncSetAttribute((const void*)hodge_main, hipFuncAttributeMaxDynamicSharedMemorySize, (int)smem);
  hodge_main<<<512, 256, smem, stream>>>(x, mask, e_i, e_j, t_ij, t_jk, t_ik, lsc,
                                         Wg, bg, Wv0, bv0, We, be, Wout, bout,
                                         g1, b1, g2, b2, Wf1, bf1, Wf2, bf2, ws, outp);
}